// NDT_82162724372480
// MI455X (gfx1250) — compile-verified
//
#include <hip/hip_runtime.h>

// Neural decision forest forward: 131072 units x 1023 logits each.
// Bandwidth-bound (536 MB @ 23.3 TB/s ~ 23 us floor). One wave32 per unit.
// CDNA5 paths: async global->LDS staging + s_wait_asynccnt, v_tanh_f32 sigmoid.

#define NF 1023
#define SLAB_STRIDE 1032          // 1023 data + shift-by-1 + pad, 16B-aligned slabs
#define WAVES_PER_BLOCK 8
#define THREADS (WAVES_PER_BLOCK * 32)

#if defined(__HIP_DEVICE_COMPILE__) && \
    __has_builtin(__builtin_amdgcn_global_load_async_to_lds_b32) && \
    __has_builtin(__builtin_amdgcn_s_wait_asynccnt)
#define USE_ASYNC_LDS 1
typedef __attribute__((address_space(1))) int g_i32;   // global (AS1)
typedef __attribute__((address_space(3))) int l_i32;   // LDS    (AS3)
#else
#define USE_ASYNC_LDS 0
#endif

// Returns tanh(x/2) == 2*sigmoid(x) - 1.  sigmoid(x) = 0.5 + 0.5*t,
// 1 - sigmoid(x) = 0.5 - 0.5*t.  One transcendental per fork on gfx1250.
__device__ __forceinline__ float sig_t(float x) {
#if defined(__HIP_DEVICE_COMPILE__) && __has_builtin(__builtin_amdgcn_tanhf)
  return __builtin_amdgcn_tanhf(0.5f * x);
#elif defined(__HIP_DEVICE_COMPILE__) && __has_builtin(__builtin_amdgcn_tanh_f32)
  return __builtin_amdgcn_tanh_f32(0.5f * x);
#else
  return 2.0f / (1.0f + __expf(-x)) - 1.0f;   // exp + rcp fallback
#endif
}

__global__ __launch_bounds__(THREADS) void ndt_forest_kernel(
    const float* __restrict__ x, float* __restrict__ out, int num_units) {
  __shared__ float lds[WAVES_PER_BLOCK][SLAB_STRIDE];

  const int lane = threadIdx.x & 31;
  const int wave = threadIdx.x >> 5;
  const int unit = blockIdx.x * WAVES_PER_BLOCK + wave;
  if (unit >= num_units) return;   // wave-uniform guard

  const float* __restrict__ src = x + (size_t)unit * NF;
  float* slab = &lds[wave][0];     // this wave's private slab; no barriers needed

  // ---- Stage 1023 logits into LDS, shifted by +1 so level i lives at 2^i ----
#if USE_ASYNC_LDS
#pragma unroll
  for (int i = 0; i < 32; ++i) {
    const int idx = i * 32 + lane;
    if (idx < NF) {
      __builtin_amdgcn_global_load_async_to_lds_b32(
          (g_i32*)(src + idx), (l_i32*)(slab + 1 + idx), 0, 0);
    }
  }
  __builtin_amdgcn_s_wait_asynccnt(0);
#if __has_builtin(__builtin_amdgcn_wave_barrier)
  __builtin_amdgcn_wave_barrier();   // scheduling fence: keep LDS reads below wait
#endif
#else
  for (int idx = lane; idx < NF; idx += 32) slab[1 + idx] = src[idx];
  // same-wave DS ops are processed in order; reads below see these stores
#endif

  // ---- Levels 0..4: every lane walks its ancestor path (31 shared nodes) ----
  float P = 1.0f;
  int node = 0;
#pragma unroll
  for (int i = 0; i < 5; ++i) {
    const int bit = (lane >> (4 - i)) & 1;
    const float t = sig_t(slab[(1 << i) + node]);
    const float h = 0.5f * P;
    const float q = h * t;
    P = bit ? (h + q) : (h - q);
    node = (node << 1) | bit;
  }

  // ---- Levels 5..8: lane l owns subtree of level-5 node l, pure registers ----
  float p[16];
  p[0] = P;
#pragma unroll
  for (int d = 0; d < 4; ++d) {
    const int n = 1 << d;
    const int base = (1 << (5 + d)) + (lane << d);
#pragma unroll
    for (int k = n - 1; k >= 0; --k) {   // downward: no overwrite hazard
      const float t = sig_t(slab[base + k]);
      const float h = 0.5f * p[k];
      const float q = h * t;
      p[2 * k]     = h - q;              // left child:  P*(1-sigmoid)
      p[2 * k + 1] = h + q;              // right child: P*sigmoid
    }
  }

  // ---- Level 9: dot with sigmoid of the 512 deepest forks (16 per lane) ----
  float acc = 0.0f;
  const int base9 = 512 + (lane << 4);
#pragma unroll
  for (int k = 0; k < 16; ++k) {
    const float t = sig_t(slab[base9 + k]);
    acc = fmaf(p[k], fmaf(0.5f, t, 0.5f), acc);   // p * sigmoid
  }

  // ---- Wave32 reduction ----
#pragma unroll
  for (int off = 16; off > 0; off >>= 1) acc += __shfl_xor(acc, off, 32);
  if (lane == 0) out[unit] = acc;
}

extern "C" void kernel_launch(void* const* d_in, const int* in_sizes, int n_in,
                              void* d_out, int out_size, void* d_ws, size_t ws_size,
                              hipStream_t stream) {
  (void)n_in; (void)out_size; (void)d_ws; (void)ws_size;
  const float* x = (const float*)d_in[0];
  float* out = (float*)d_out;
  const int num_units = in_sizes[0] / NF;                      // 2048*64 = 131072
  const int blocks = (num_units + WAVES_PER_BLOCK - 1) / WAVES_PER_BLOCK;
  ndt_forest_kernel<<<blocks, THREADS, 0, stream>>>(x, out, num_units);
}